// SimilarityHead_8873402434333
// MI455X (gfx1250) — compile-verified
//
#include <hip/hip_runtime.h>
#include <math.h>

typedef float v2f __attribute__((ext_vector_type(2)));
typedef float v8f __attribute__((ext_vector_type(8)));

#define BATCH 8
#define CHAN  256
#define HDIM  128
#define WDIM  128
#define HW    (HDIM * WDIM)        // 16384
#define KWIN  7
#define PADR  3                    // (k-1)/2
#define TILE  32                   // 32x32 output block per workgroup
#define HALO  (TILE + 2 * PADR)    // 38
#define YSTR  42                   // LDS row stride for y halo (8 rows -> +16 banks)
#define HSSTR 48                   // LDS row stride for horiz sums (2 rows -> +32 banks)
#define NLD   12                   // ceil(HALO*HALO / 128)
#define YDUMP (HALO * YSTR)        // 128 per-thread dump slots after the halo

// ---------------------------------------------------------------------------
// Kernel 1: per-pixel inverse channel norm  rn = |x|>0 ? 1/|x| : 0
// ---------------------------------------------------------------------------
__global__ void __launch_bounds__(256) cos_rnorm_kernel(const float* __restrict__ x,
                                                        float* __restrict__ rn) {
    int p = blockIdx.x * 256 + threadIdx.x;          // pixel id in [0, BATCH*HW)
    if (p >= BATCH * HW) return;
    int b  = p >> 14;                                // p / HW
    int hw = p & (HW - 1);
    const float* xp = x + (size_t)b * CHAN * HW + hw;
    float s = 0.0f;
#pragma unroll 8
    for (int c = 0; c < CHAN; ++c) {
        float v = xp[(size_t)c * HW];
        s = fmaf(v, v, s);
    }
    rn[p] = (s > 0.0f) ? (1.0f / sqrtf(s)) : 0.0f;
}

// ---------------------------------------------------------------------------
// Kernel 2: out(p) = (1/49) * sum_c  y_c(p) * box7x7(y_c)(p),  y = x * rn.
// Per channel:  d(16x16) = Band(16x24) x HS(24x16) via 6 chained
// V_WMMA_F32_16X16X4_F32, then acc += d * y(center).
// All index maps are channel-invariant registers (unsigned 32-bit element
// offsets + uniform scalar plane pointer -> saddr-form global loads);
// invalid staging elements go to a dump LDS slot (branch-free P1/prefetch);
// next channel's globals are prefetched into registers before the WMMA phase.
// Block: 128 threads = 4 waves; wave (ty,tx) owns a 16x16 output sub-tile.
// ---------------------------------------------------------------------------
__global__ void __launch_bounds__(128) cos_mean_kernel(const float* __restrict__ x,
                                                       const float* __restrict__ rn,
                                                       float* __restrict__ out) {
    __shared__ float ys[HALO * YSTR + 128]; // normalized halo + per-thread dump
    __shared__ float hs[HALO * HSSTR];      // horizontal 7-tap sums

    const int b   = blockIdx.z;
    const int h0  = blockIdx.y * TILE;
    const int w0  = blockIdx.x * TILE;
    const int tid  = threadIdx.x;                    // 0..127
    const int wave = tid >> 5;
    const int lane = tid & 31;
    const int ty = wave >> 1, tx = wave & 1;         // 2x2 sub-tiles of 16x16
    const int m    = lane & 15;                      // M (A rows) == N (B cols)
    const int half = lane >> 4;

    const float* xb  = x  + (size_t)b * CHAN * HW;
    const float* rnb = rn + b * HW;

    // ---- P1 map (channel-invariant): 12 staged elements per thread.
    // Invalid elements: goff=0 (safe read), rmul=0 (value killed),
    // laddr=dump slot (store harmless).  -> completely branch-free per channel.
    unsigned goff [NLD];   // 32-bit element offset within a channel plane
    int      laddr[NLD];
    float    rmul [NLD];
#pragma unroll
    for (int j = 0; j < NLD; ++j) {
        int idx = tid + 128 * j;
        int r  = idx / HALO, cx = idx - r * HALO;
        int gh = h0 - PADR + r, gw = w0 - PADR + cx;
        bool in = ((unsigned)gh < HDIM) && ((unsigned)gw < WDIM);
        bool ok = (idx < HALO * HALO) && in;
        goff [j] = ok ? (unsigned)(gh * WDIM + gw) : 0u;
        laddr[j] = ok ? (r * YSTR + cx) : (YDUMP + tid);
        rmul [j] = ok ? rnb[gh * WDIM + gw] : 0.0f;
        if (idx < HALO * HALO && !in)                // zero-pad cells: constant
            ys[r * YSTR + cx] = 0.0f;                // across channels, write once
    }

    // ---- P2 map: sliding-window tasks (152 total; second task iff tid<24) ----
    const int  r0 = tid >> 2,          c0 = (tid & 3) * 8;
    const int  ya = r0 * YSTR + c0,    ha = r0 * HSSTR + c0;
    const bool has2 = (tid < HALO * 4 - 128);
    const int  r1 = (tid + 128) >> 2,  c1 = ((tid + 128) & 3) * 8;
    const int  yb = r1 * YSTR + c1,    hb = r1 * HSSTR + c1;

    // ---- P3 maps: band constants, B-operand and center LDS offsets ----
    // f32 16x4 A layout: VGPR v, lane-half h -> K = k0 + v + 2h; B mirrored.
    float aband[12];
    int   boff [12];
#pragma unroll
    for (int k = 0; k < 6; ++k)
#pragma unroll
        for (int v = 0; v < 2; ++v) {
            int K = 4 * k + v + 2 * half;
            aband[2 * k + v] = (K >= m && K <= m + 6) ? 1.0f : 0.0f;
            int rr = 16 * ty + K; if (rr > HALO - 1) rr = HALO - 1;  // A==0 there
            boff[2 * k + v] = rr * HSSTR + 16 * tx + m;
        }
    int coff[8];
#pragma unroll
    for (int r = 0; r < 8; ++r)
        coff[r] = (16 * ty + r + 8 * half + PADR) * YSTR + (16 * tx + m + PADR);

    // ---- prefetch channel 0 ----
    float pf[NLD];
#pragma unroll
    for (int j = 0; j < NLD; ++j)
        pf[j] = xb[goff[j]] * rmul[j];

    v8f acc = {0.f, 0.f, 0.f, 0.f, 0.f, 0.f, 0.f, 0.f};

    // Uniform plane pointer for the *next* channel (scalar add per iteration;
    // loads become saddr-base + 32-bit unsigned voffset, no per-lane 64-bit math).
    const float* xc = xb + HW;

    for (int c = 0; c < CHAN; ++c) {
        // P1: register -> LDS (unconditional; invalid -> dump slot)
#pragma unroll
        for (int j = 0; j < NLD; ++j)
            ys[laddr[j]] = pf[j];
        __syncthreads();

        // P2: horizontal 7-tap sums (sliding window, 8 outputs per task)
        {
            const float* yr = &ys[ya];
            float s = yr[0] + yr[1] + yr[2] + yr[3] + yr[4] + yr[5] + yr[6];
            float* hr = &hs[ha];
            hr[0] = s;
#pragma unroll
            for (int j = 1; j < 8; ++j) { s += yr[j + 6] - yr[j - 1]; hr[j] = s; }
        }
        if (has2) {
            const float* yr = &ys[yb];
            float s = yr[0] + yr[1] + yr[2] + yr[3] + yr[4] + yr[5] + yr[6];
            float* hr = &hs[hb];
            hr[0] = s;
#pragma unroll
            for (int j = 1; j < 8; ++j) { s += yr[j + 6] - yr[j - 1]; hr[j] = s; }
        }
        __syncthreads();

        // Prefetch next channel's globals (overlaps with the WMMA phase).
        // Last iteration re-reads channel CHAN-1 (L2 hit) to keep the body uniform.
#pragma unroll
        for (int j = 0; j < NLD; ++j)
            pf[j] = xc[goff[j]] * rmul[j];
        if (c < CHAN - 2) xc += HW;                  // uniform scalar advance

        // P3: batch-load B operands and center values, then
        //     d = Band(16x24) x HS(24x16);  acc += d * y(center).
        float bl[12];
#pragma unroll
        for (int i = 0; i < 12; ++i) bl[i] = hs[boff[i]];
        float cl[8];
#pragma unroll
        for (int r = 0; r < 8; ++r) cl[r] = ys[coff[r]];

        v8f d = {0.f, 0.f, 0.f, 0.f, 0.f, 0.f, 0.f, 0.f};
#if __has_builtin(__builtin_amdgcn_wmma_f32_16x16x4_f32)
#pragma unroll
        for (int k = 0; k < 6; ++k) {
            v2f av, bv;
            av[0] = aband[2 * k];  av[1] = aband[2 * k + 1];
            bv[0] = bl[2 * k];     bv[1] = bl[2 * k + 1];
            d = __builtin_amdgcn_wmma_f32_16x16x4_f32(
                false, av, false, bv, (short)0, d, false, false);
        }
#else
#pragma unroll
        for (int r = 0; r < 8; ++r) {
            int M = r + 8 * half;
            float s = 0.0f;
#pragma unroll
            for (int j = 0; j < 7; ++j)
                s += hs[(16 * ty + M + j) * HSSTR + 16 * tx + m];
            d[r] = s;
        }
#endif
#pragma unroll
        for (int r = 0; r < 8; ++r)
            acc[r] = fmaf(d[r], cl[r], acc[r]);
        __syncthreads();
    }

    // C/D layout: VGPR r, lane (m, half) -> element (M = r + 8*half, N = m).
    const float inv = 1.0f / (float)(KWIN * KWIN);
#pragma unroll
    for (int r = 0; r < 8; ++r) {
        int gh = h0 + 16 * ty + r + 8 * half;
        int gw = w0 + 16 * tx + m;
        out[b * HW + gh * WDIM + gw] = acc[r] * inv;
    }
}

// ---------------------------------------------------------------------------
extern "C" void kernel_launch(void* const* d_in, const int* in_sizes, int n_in,
                              void* d_out, int out_size, void* d_ws, size_t ws_size,
                              hipStream_t stream) {
    const float* x = (const float*)d_in[0];
    // d_in[1] is k == 7 (python scalar); kernel is specialized for k=7, pad=3.
    float* rnorm = (float*)d_ws;               // BATCH*HW floats = 512 KB scratch
    float* out   = (float*)d_out;              // (8,1,128,128) float32

    int npix = BATCH * HW;                      // 131072
    cos_rnorm_kernel<<<(npix + 255) / 256, 256, 0, stream>>>(x, rnorm);

    dim3 grid(WDIM / TILE, HDIM / TILE, BATCH); // (4, 4, 8)
    cos_mean_kernel<<<grid, 128, 0, stream>>>(x, rnorm, out);
}